// Pooling_37572373906119
// MI455X (gfx1250) — compile-verified
//
#include <hip/hip_runtime.h>
#include <math.h>

#define T_LEN   16384
#define D_IN    512
#define N_OUT   512
#define K_TOT   4096
#define M_BLK   64                 // output rows per block (4 M-tiles)
#define CHUNK   128                // feature columns staged per phase
#define LDS_STRIDE 132             // 128 + 4 pad -> conflict-free wmma A reads
#define LDS_ROWS   72              // 71 staged x rows + 1 zero row (r=71)

typedef float v2f __attribute__((ext_vector_type(2)));
typedef float v8f __attribute__((ext_vector_type(8)));

__device__ __forceinline__ float cos8(int m) {
  // cos(2*pi*m/8), m in [0,8)
  const float R = 0.70710678118654752440f;
  float v;
  switch (m & 7) {
    case 0: v =  1.0f; break;
    case 1: v =  R;    break;
    case 2: v =  0.0f; break;
    case 3: v = -R;    break;
    case 4: v = -1.0f; break;
    case 5: v = -R;    break;
    case 6: v =  0.0f; break;
    default: v = R;    break;
  }
  return v;
}

// ---------------------------------------------------------------------------
// Kernel 1: fold the 8x8 real-FFT cosine matrix into W, and repack into a
// WMMA-friendly K-pair layout:  Wpp[((k>>1)*512 + n)*2 + (k&1)] = W'[k][n]
// where W'[j*512+d][n] = sum_kk cos(2*pi*kk*j/8) * W[(kk*512+d)*512 + n].
// ---------------------------------------------------------------------------
__global__ __launch_bounds__(256) void fold_w_kernel(
    const float* __restrict__ W, float* __restrict__ Wpp)
{
  int idx = blockIdx.x * 256 + threadIdx.x;        // 0 .. 4096*512-1
  if (idx >= K_TOT * N_OUT) return;
  int k = idx >> 9;                                 // 0..4095  (= j*512 + d)
  int n = idx & 511;
  int j = k >> 9;                                   // window position 0..7
  int d = k & 511;                                  // feature 0..511
  float acc = 0.0f;
#pragma unroll
  for (int kk = 0; kk < 8; ++kk) {
    acc += cos8((kk * j) & 7) * W[((size_t)(kk * 512 + d)) * N_OUT + n];
  }
  Wpp[(((size_t)(k >> 1)) * N_OUT + n) * 2 + (k & 1)] = acc;
}

// ---------------------------------------------------------------------------
// Kernel 2: coarse = mean_k Re(FFT_8) == window row 0 == x[max(0, i-3)]
// ---------------------------------------------------------------------------
__global__ __launch_bounds__(256) void coarse_kernel(
    const float* __restrict__ x, float* __restrict__ coarse)
{
  int idx = blockIdx.x * 256 + threadIdx.x;        // float4 index
  if (idx >= T_LEN * (D_IN / 4)) return;
  int i = idx >> 7;                                 // row (128 float4 per row)
  int c = idx & 127;
  int src = (i - 3) > 0 ? (i - 3) : 0;
  const float4* xr = (const float4*)(x + (size_t)src * D_IN);
  ((float4*)(coarse + (size_t)i * D_IN))[c] = xr[c];
}

// ---------------------------------------------------------------------------
// Kernel 3: fine = GELU( Xwin[16384 x 4096] @ W'[4096 x 512] + b )
// One block: M=64 rows x N=512 cols, 256 threads = 8 waves. Each wave owns a
// 64-column slab and all 4 M-tiles -> each B fragment feeds 4 WMMAs, cutting
// L2 weight traffic 4x vs an M=16 tile. x is staged through LDS in 4 phases
// of 128 feature columns to stay under the 64 KB static-LDS limit.
// ---------------------------------------------------------------------------
__global__ __launch_bounds__(256) void gemm_gelu_kernel(
    const float* __restrict__ x, const float* __restrict__ Wpp,
    const float* __restrict__ bias, float* __restrict__ fine)
{
  __shared__ float xs[LDS_ROWS * LDS_STRIDE];      // 38 KB

  const int tid = threadIdx.x;
  const int i0  = blockIdx.x * M_BLK;               // first output row

  const int lane = tid & 31;
  const int wave = tid >> 5;
  const int m    = lane & 15;        // row within 16x16 tile (A layout)
  const int lh   = lane >> 4;        // lane half: K offset 0 / 2
  const int koff = lh * 2;
  const int n0   = wave * 64;        // this wave's column slab
  const int nl   = lane & 15;        // column within tile (B/C layout)

  // LDS row offset per (m-tile, window slot j); exact left-edge reorder and
  // right-edge zero handling; row 71 is the staged zero row.
  int rbase[4][8];
#pragma unroll
  for (int mt = 0; mt < 4; ++mt) {
    const int i = i0 + mt * 16 + m;
    const int start = (i - 3) > 0 ? (i - 3) : 0;
    const int Lw    = ((i + 5) < T_LEN ? (i + 5) : T_LEN) - start;
#pragma unroll
    for (int j = 0; j < 8; ++j) {
      const int r = (j < Lw) ? (start + j - (i0 - 3)) : 71;
      rbase[mt][j] = r * LDS_STRIDE;
    }
  }

  v8f acc[4][4];
#pragma unroll
  for (int mt = 0; mt < 4; ++mt)
#pragma unroll
    for (int nt = 0; nt < 4; ++nt)
#pragma unroll
      for (int e = 0; e < 8; ++e) acc[mt][nt][e] = 0.0f;

  for (int ph = 0; ph < 4; ++ph) {
    const int c0 = ph * CHUNK;
    __syncthreads();               // previous phase's reads complete

    // stage 71 x-rows [i0-3 .. i0+67], cols [c0, c0+128); row 71 stays zero
    for (int t = tid; t < LDS_ROWS * (CHUNK / 4); t += 256) {   // 9 iters
      int r  = t >> 5;
      int c4 = t & 31;
      int g  = i0 - 3 + r;
      float4 v = make_float4(0.f, 0.f, 0.f, 0.f);
      if (r < 71 && g >= 0 && g < T_LEN)
        v = ((const float4*)(x + (size_t)g * D_IN + c0))[c4];
      *(float4*)&xs[r * LDS_STRIDE + c4 * 4] = v;
    }
    __syncthreads();

#pragma unroll
    for (int j = 0; j < 8; ++j) {
      // B pair index p = j*256 + (c0+dd)/2 + lh ; float offset = (p*512+n)*2
      const float* __restrict__ bbase =
          Wpp + ((size_t)(j * 256 + (c0 >> 1) + lh)) * 1024 +
          (size_t)(n0 + nl) * 2;

#pragma unroll 2
      for (int dd = 0; dd < CHUNK; dd += 4) {
        const float* bp = bbase + (size_t)dd * 512;  // +8 KiB per step
        v2f b0 = *(const v2f*)(bp);
        v2f b1 = *(const v2f*)(bp + 32);
        v2f b2 = *(const v2f*)(bp + 64);
        v2f b3 = *(const v2f*)(bp + 96);
#pragma unroll
        for (int mt = 0; mt < 4; ++mt) {
          v2f a = *(const v2f*)(xs + rbase[mt][j] + dd + koff);
          acc[mt][0] = __builtin_amdgcn_wmma_f32_16x16x4_f32(
              false, a, false, b0, (short)0, acc[mt][0], false, false);
          acc[mt][1] = __builtin_amdgcn_wmma_f32_16x16x4_f32(
              false, a, false, b1, (short)0, acc[mt][1], false, false);
          acc[mt][2] = __builtin_amdgcn_wmma_f32_16x16x4_f32(
              false, a, false, b2, (short)0, acc[mt][2], false, false);
          acc[mt][3] = __builtin_amdgcn_wmma_f32_16x16x4_f32(
              false, a, false, b3, (short)0, acc[mt][3], false, false);
        }
      }
    }
  }

  // ---- epilogue: bias + exact GELU, store -------------------------------
  // C/D layout: VGPR v holds row (v + lh*8), column nl.
#pragma unroll
  for (int nt = 0; nt < 4; ++nt) {
    const int colg = n0 + nt * 16 + nl;
    const float bv = bias[colg];
#pragma unroll
    for (int mt = 0; mt < 4; ++mt) {
#pragma unroll
      for (int v = 0; v < 8; ++v) {
        const int row = mt * 16 + v + lh * 8;
        float val = acc[mt][nt][v] + bv;
        val = 0.5f * val * (1.0f + erff(val * 0.70710678118654752440f));
        fine[(size_t)(i0 + row) * N_OUT + colg] = val;
      }
    }
  }
}

// ---------------------------------------------------------------------------
extern "C" void kernel_launch(void* const* d_in, const int* in_sizes, int n_in,
                              void* d_out, int out_size, void* d_ws, size_t ws_size,
                              hipStream_t stream) {
  (void)in_sizes; (void)n_in; (void)out_size; (void)ws_size;
  const float* x  = (const float*)d_in[0];   // [1,16384,512]
  const float* W  = (const float*)d_in[1];   // [4096,512]
  const float* b  = (const float*)d_in[2];   // [512]
  float* coarse = (float*)d_out;                              // [16384,512]
  float* fine   = (float*)d_out + (size_t)T_LEN * D_IN;       // [16384,512]
  float* Wpp    = (float*)d_ws;  // 8 MiB folded+repacked weights

  fold_w_kernel <<<(K_TOT * N_OUT) / 256, 256, 0, stream>>>(W, Wpp);
  coarse_kernel <<<(T_LEN * D_IN / 4) / 256, 256, 0, stream>>>(x, coarse);
  gemm_gelu_kernel<<<T_LEN / M_BLK, 256, 0, stream>>>(x, Wpp, b, fine);
}